// DecoderDecodeLayer_62680752718087
// MI455X (gfx1250) — compile-verified
//
#include <hip/hip_runtime.h>
#include <math.h>

#define H      2048
#define NH     16
#define NKV    2
#define HD     256
#define RD     64
#define MAXSEQ 8192
#define DFF    8192
#define EPSV   1e-6f
#define SCALEV 0.0625f   // 256^-0.5
#define GROUPS 8

typedef __attribute__((ext_vector_type(2))) float v2f;
typedef __attribute__((ext_vector_type(8))) float v8f;

// ---------------- RMSNorm (one block, n multiple of 256) ----------------
__global__ __launch_bounds__(256) void rmsnorm_kernel(const float* __restrict__ x,
                                                      const float* __restrict__ w,
                                                      float* __restrict__ y, int n) {
    __shared__ float red[256];
    int t = threadIdx.x;
    float s = 0.f;
    for (int i = t; i < n; i += 256) { float v = x[i]; s += v * v; }
    red[t] = s; __syncthreads();
    for (int o = 128; o > 0; o >>= 1) { if (t < o) red[t] += red[t + o]; __syncthreads(); }
    float rs = rsqrtf(red[0] / (float)n + EPSV);
    for (int i = t; i < n; i += 256) y[i] = x[i] * rs * (1.f + w[i]);
}

// ---------------- GEMV: y[r] = dot(W[r,:], x) (+res), wave per row ------
__global__ __launch_bounds__(256) void gemv_kernel(const float* __restrict__ W,
                                                   const float* __restrict__ x,
                                                   const float* __restrict__ res,
                                                   float* __restrict__ y, int K) {
    int lane = threadIdx.x & 31;
    int wid  = threadIdx.x >> 5;
    int row  = blockIdx.x * 8 + wid;
    const float* wr = W + (size_t)row * K;
    float s = 0.f;
    for (int k = lane * 4; k < K; k += 128) {
        float4 wv = *(const float4*)(wr + k);
        float4 xv = *(const float4*)(x  + k);
        s += wv.x * xv.x + wv.y * xv.y + wv.z * xv.z + wv.w * xv.w;
    }
#pragma unroll
    for (int o = 16; o > 0; o >>= 1) s += __shfl_xor(s, o, 32);
    if (lane == 0) y[row] = s + (res ? res[row] : 0.f);
}

// ------- per-head q/k RMSNorm + RoPE + KV cache row write + zero attn ---
// q is written into a zero-padded [NKV][16][256] layout (rows 8..15 = 0)
// so the WMMA scores kernel needs no exec-masked A loads.
__global__ __launch_bounds__(256) void qknorm_rope_kernel(
    const float* __restrict__ qg, const float* __restrict__ kproj,
    const float* __restrict__ vproj, const float* __restrict__ qnw,
    const float* __restrict__ knw, const float* __restrict__ cosp,
    const float* __restrict__ sinp, const float* __restrict__ position,
    float* __restrict__ qnp, float* __restrict__ kcache_out,
    float* __restrict__ vcache_out, float* __restrict__ attn_zero) {
    __shared__ float red[256];
    __shared__ float buf[256];
    int b = blockIdx.x, t = threadIdx.x;
    float val; const float* nw;
    if (b < 16) { val = qg[b * 512 + t];          nw = qnw; }
    else        { val = kproj[(b - 16) * 256 + t]; nw = knw; }
    red[t] = val * val; __syncthreads();
    for (int o = 128; o > 0; o >>= 1) { if (t < o) red[t] += red[t + o]; __syncthreads(); }
    float rs = rsqrtf(red[0] / 256.0f + EPSV);
    float nv = val * rs * (1.f + nw[t]);
    buf[t] = nv; __syncthreads();
    float outv = nv;
    if (t < RD) {
        float rh = (t < 32) ? -buf[t + 32] : buf[t - 32];
        outv = nv * cosp[t] + rh * sinp[t];
    }
    if (b < 16) {
        int g  = b >> 3;           // kv group
        int hl = b & 7;            // head within group
        qnp[(size_t)g * 16 * HD + hl * HD + t]       = outv;  // real row
        qnp[(size_t)g * 16 * HD + (8 + hl) * HD + t] = 0.f;   // zero pad row
        attn_zero[b * 256 + t] = 0.f;
    } else {
        int j = b - 16;
        int pos = (int)(position[0] + 0.5f);
        kcache_out[(size_t)j * MAXSEQ * HD + (size_t)pos * HD + t] = outv;
        vcache_out[(size_t)j * MAXSEQ * HD + (size_t)pos * HD + t] = vproj[j * 256 + t];
    }
}

// ------- attention scores via V_WMMA_F32_16X16X4_F32, 4 waves/block -----
// A (16x4 f32): lane l holds M = l%16, K = 2*(l/16)+{0,1} (pre-zero-padded)
// B (4x16 f32): lane l holds N = l%16, K = 2*(l/16)+{0,1}
// C/D (16x16 f32): VGPR r, lane l: M = r + 8*(l/16), N = l%16
__global__ __launch_bounds__(128) void attn_scores_wmma(
    const float* __restrict__ qnp, const float* __restrict__ knew,
    float* __restrict__ scores) {
    int lane  = threadIdx.x & 31;
    int wave  = threadIdx.x >> 5;
    int s0    = (blockIdx.x * 4 + wave) * 16;
    int g     = blockIdx.y;
    int mrow  = lane & 15;
    int khalf = (lane >> 4) * 2;   // 0 or 2
    const float* qbase = qnp + (size_t)g * 16 * HD + mrow * HD + khalf;
    const float* kbase = knew + (size_t)g * MAXSEQ * HD + (size_t)s0 * HD + mrow * HD + khalf;
    v8f c = {0.f, 0.f, 0.f, 0.f, 0.f, 0.f, 0.f, 0.f};
#pragma unroll 8
    for (int kc = 0; kc < HD; kc += 4) {
        v2f a = *(const v2f*)(qbase + kc);
        v2f b = *(const v2f*)(kbase + kc);
        c = __builtin_amdgcn_wmma_f32_16x16x4_f32(false, a, false, b,
                                                  (short)0, c, false, false);
    }
    if (lane < 16) {
#pragma unroll
        for (int r = 0; r < 8; ++r)
            scores[(size_t)(g * 8 + r) * MAXSEQ + s0 + lane] = c[r] * SCALEV;
    }
}

// ---------------- softmax per head with causal -10000 mask --------------
__global__ __launch_bounds__(256) void softmax_kernel(float* __restrict__ scores,
                                                      const float* __restrict__ position) {
    __shared__ float red[256];
    int h = blockIdx.x, t = threadIdx.x;
    int pos = (int)(position[0] + 0.5f);
    float* row = scores + (size_t)h * MAXSEQ;
    float mx = -1e30f;
    for (int s = t; s <= pos; s += 256) mx = fmaxf(mx, row[s]);
    red[t] = mx; __syncthreads();
    for (int o = 128; o > 0; o >>= 1) { if (t < o) red[t] = fmaxf(red[t], red[t + o]); __syncthreads(); }
    mx = red[0]; __syncthreads();
    float sum = 0.f;
    for (int s = t; s <= pos; s += 256) sum += expf(row[s] - mx);
    red[t] = sum; __syncthreads();
    for (int o = 128; o > 0; o >>= 1) { if (t < o) red[t] += red[t + o]; __syncthreads(); }
    float inv = 1.f / red[0];
    for (int s = t; s < MAXSEQ; s += 256)
        row[s] = (s <= pos) ? expf(row[s] - mx) * inv : 0.f;
}

// -------- weighted V: grid (seq_chunks, d_tiles, groups), atomic acc ----
__global__ __launch_bounds__(256) void attn_out_kernel(
    const float* __restrict__ wts, const float* __restrict__ vnew,
    const float* __restrict__ position, float* __restrict__ attn) {
    int pos = (int)(position[0] + 0.5f);
    int s_begin = blockIdx.x * 256;
    if (s_begin > pos) return;
    int s_end = min(pos, s_begin + 255);
    int g  = blockIdx.z;
    int d  = blockIdx.y * 64 + (threadIdx.x & 63);
    int hp = threadIdx.x >> 6;            // head pair 0..3
    const float* vb = vnew + (size_t)g * MAXSEQ * HD;
    const float* w0 = wts + (size_t)(g * 8 + hp * 2) * MAXSEQ;
    const float* w1 = w0 + MAXSEQ;
    float a0 = 0.f, a1 = 0.f;
    for (int s = s_begin; s <= s_end; ++s) {
        float vv = vb[(size_t)s * HD + d];
        a0 += w0[s] * vv;
        a1 += w1[s] * vv;
    }
    atomicAdd(&attn[(g * 8 + hp * 2)     * HD + d], a0);
    atomicAdd(&attn[(g * 8 + hp * 2 + 1) * HD + d], a1);
}

// ---------------- apply sigmoid gate -----------------------------------
__global__ __launch_bounds__(256) void oin_kernel(const float* __restrict__ attn,
                                                  const float* __restrict__ qg,
                                                  float* __restrict__ oin) {
    int i = blockIdx.x * 256 + threadIdx.x;   // < 4096
    int h = i >> 8, d = i & 255;
    float gate = qg[h * 512 + 256 + d];
    oin[i] = attn[i] / (1.f + expf(-gate));
}

// ------- fused gate/up GEMV: t[r] = silu(gw.x) * (uw.x) -----------------
__global__ __launch_bounds__(256) void mlp_gateup_kernel(
    const float* __restrict__ gw, const float* __restrict__ uw,
    const float* __restrict__ x, float* __restrict__ t_out, int K) {
    int lane = threadIdx.x & 31;
    int wid  = threadIdx.x >> 5;
    int row  = blockIdx.x * 8 + wid;
    const float* gr = gw + (size_t)row * K;
    const float* ur = uw + (size_t)row * K;
    float sg = 0.f, su = 0.f;
    for (int k = lane * 4; k < K; k += 128) {
        float4 xv = *(const float4*)(x  + k);
        float4 gv = *(const float4*)(gr + k);
        float4 uv = *(const float4*)(ur + k);
        sg += gv.x * xv.x + gv.y * xv.y + gv.z * xv.z + gv.w * xv.w;
        su += uv.x * xv.x + uv.y * xv.y + uv.z * xv.z + uv.w * xv.w;
    }
#pragma unroll
    for (int o = 16; o > 0; o >>= 1) { sg += __shfl_xor(sg, o, 32); su += __shfl_xor(su, o, 32); }
    if (lane == 0) {
        float sil = sg / (1.f + expf(-sg));
        t_out[row] = sil * su;
    }
}

extern "C" void kernel_launch(void* const* d_in, const int* in_sizes, int n_in,
                              void* d_out, int out_size, void* d_ws, size_t ws_size,
                              hipStream_t stream) {
    const float* hidden_in    = (const float*)d_in[0];
    const float* position     = (const float*)d_in[1];
    const float* cosp         = (const float*)d_in[2];
    const float* sinp         = (const float*)d_in[3];
    const float* k_cache      = (const float*)d_in[4];
    const float* v_cache      = (const float*)d_in[5];
    const float* q_proj_w     = (const float*)d_in[6];
    const float* k_proj_w     = (const float*)d_in[7];
    const float* v_proj_w     = (const float*)d_in[8];
    const float* o_proj_w     = (const float*)d_in[9];
    const float* q_norm_w     = (const float*)d_in[10];
    const float* k_norm_w     = (const float*)d_in[11];
    const float* input_norm_w = (const float*)d_in[12];
    const float* post_norm_w  = (const float*)d_in[13];
    const float* gate_w       = (const float*)d_in[14];
    const float* up_w         = (const float*)d_in[15];
    const float* down_w       = (const float*)d_in[16];

    float* out  = (float*)d_out;
    float* knew = out + H;                                  // new_k cache
    float* vnew = knew + (size_t)NKV * MAXSEQ * HD;         // new_v cache

    float* ws = (float*)d_ws;
    float* X      = ws;             // 2048
    float* QG     = ws + 2048;      // 8192 (16 heads x [q(256)|gate(256)])
    float* KPROJ  = ws + 10240;     // 512
    float* VPROJ  = ws + 10752;     // 512
    float* QNP    = ws + 11264;     // 8192: padded q, [NKV][16][256], rows 8..15 zero
    float* SCORES = ws + 19456;     // 16*8192 scores -> softmax weights
    float* ATTN   = SCORES + (size_t)NH * MAXSEQ;   // 4096
    float* OIN    = ATTN + 4096;    // 4096
    float* HBUF   = OIN + 4096;     // 2048
    float* M      = HBUF + 2048;    // 2048
    float* T      = M + 2048;       // 8192

    size_t cache_bytes = (size_t)NKV * MAXSEQ * HD * sizeof(float);
    hipMemcpyAsync(knew, k_cache, cache_bytes, hipMemcpyDeviceToDevice, stream);
    hipMemcpyAsync(vnew, v_cache, cache_bytes, hipMemcpyDeviceToDevice, stream);

    // x = rmsnorm(hidden_in)
    rmsnorm_kernel<<<1, 256, 0, stream>>>(hidden_in, input_norm_w, X, H);
    // projections
    gemv_kernel<<<1024, 256, 0, stream>>>(q_proj_w, X, nullptr, QG, H);
    gemv_kernel<<<64, 256, 0, stream>>>(k_proj_w, X, nullptr, KPROJ, H);
    gemv_kernel<<<64, 256, 0, stream>>>(v_proj_w, X, nullptr, VPROJ, H);
    // q/k norm + rope + cache row update + zero attn acc
    qknorm_rope_kernel<<<18, 256, 0, stream>>>(QG, KPROJ, VPROJ, q_norm_w, k_norm_w,
                                               cosp, sinp, position, QNP, knew, vnew, ATTN);
    // scores (WMMA f32 16x16x4), 4 waves per block, 1 seq-tile per wave
    attn_scores_wmma<<<dim3(MAXSEQ / 64, NKV), 128, 0, stream>>>(QNP, knew, SCORES);
    // softmax with causal mask
    softmax_kernel<<<NH, 256, 0, stream>>>(SCORES, position);
    // weighted V
    attn_out_kernel<<<dim3(MAXSEQ / 256, 4, NKV), 256, 0, stream>>>(SCORES, vnew, position, ATTN);
    // gate
    oin_kernel<<<16, 256, 0, stream>>>(ATTN, QG, OIN);
    // o_proj + residual
    gemv_kernel<<<256, 256, 0, stream>>>(o_proj_w, OIN, hidden_in, HBUF, NH * HD);
    // post norm
    rmsnorm_kernel<<<1, 256, 0, stream>>>(HBUF, post_norm_w, M, H);
    // MLP gate/up fused
    mlp_gateup_kernel<<<1024, 256, 0, stream>>>(gate_w, up_w, M, T, H);
    // down + residual -> final out
    gemv_kernel<<<256, 256, 0, stream>>>(down_w, T, HBUF, out, DFF);
}